// NativeCTCcudaLoss_11965778886958
// MI455X (gfx1250) — compile-verified
//
#include <hip/hip_runtime.h>
#include <hip/hip_bf16.h>

// CTC loss forward (matches the JAX reference: finite NEG = -1e30 sentinel,
// zero_infinity, reduction='mean' with per-sample /target_length).
//
// log_probs [T=128,B=64,C=6625] f32, targets [B,S=32] i32, input_lengths [B],
// target_lengths [B]  ->  scalar f32.
//
// One workgroup (2 wave32 waves) per batch element. Emissions for the 65
// extended-label states are pre-gathered into LDS — on gfx1250 via the
// CDNA5 async-to-LDS path (GLOBAL_LOAD_ASYNC_TO_LDS_B32 + s_wait_asynccnt)
// — then the 128-step alpha recurrence runs entirely out of LDS with one
// barrier per step. Deterministic tree reduction in a second kernel.

#define CTC_NEG  (-1e30f)
#define CTC_TMAX 128
#define CTC_L    65          // 2*S+1 with S=32
#define CTC_S    32

#if defined(__HIP_DEVICE_COMPILE__) && __has_builtin(__builtin_amdgcn_global_load_async_to_lds_b32)
#define CTC_ASYNC_GATHER 1
#else
#define CTC_ASYNC_GATHER 0
#endif

__device__ __forceinline__ float ctc_lae(float a, float b) {
    // logaddexp, NaN-free with the finite NEG sentinel:
    //   both NEG -> NEG + log(2) (~ -1e30, matches reference math)
    //   one NEG  -> the finite one (exp(-huge) == 0)
    float mx = fmaxf(a, b);
    float mn = fminf(a, b);
    return mx + log1pf(__expf(mn - mx));
}

#if CTC_ASYNC_GATHER
typedef __attribute__((address_space(1))) int* ctc_gbl_i32p;
typedef __attribute__((address_space(3))) int* ctc_lds_i32p;

__device__ __forceinline__ void ctc_async_gather_b32(const float* gsrc, float* lds_dst) {
    // Generic LDS pointer's low 32 bits are the DS byte offset (aperture in
    // the high bits), so truncate+inttoptr to form the AS(3) pointer.
    ctc_gbl_i32p g = (ctc_gbl_i32p)(uintptr_t)gsrc;
    ctc_lds_i32p l = (ctc_lds_i32p)(uintptr_t)(unsigned)(uintptr_t)lds_dst;
    __builtin_amdgcn_global_load_async_to_lds_b32(g, l, /*imm offset*/0, /*cpol*/0);
}
__device__ __forceinline__ void ctc_wait_async0() {
#if __has_builtin(__builtin_amdgcn_s_wait_asynccnt)
    __builtin_amdgcn_s_wait_asynccnt(0);
#else
    asm volatile("s_wait_asynccnt 0x0" ::: "memory");
#endif
}
#endif

__global__ __launch_bounds__(64)
void ctc_alpha_kernel(const float* __restrict__ log_probs,   // [T,B,C]
                      const int*   __restrict__ targets,     // [B,S]
                      const int*   __restrict__ input_len,   // [B]
                      const int*   __restrict__ target_len,  // [B]
                      float*       __restrict__ per_b,       // [B] loss_b / tl_b
                      int T, int B, int C)
{
    const int b   = blockIdx.x;
    const int tid = threadIdx.x;   // 0..63, 2 waves

    __shared__ int           ext[CTC_L];                 // extended label ids
    __shared__ unsigned char skip[CTC_L];                // skip transition allowed
    __shared__ float         emit[CTC_TMAX * CTC_L];     // gathered emissions, 33 KB
    __shared__ float         abuf[2][CTC_L + 2];         // alpha double buffer + 2 NEG pads

    const int Tb = input_len[b];    // in [100, 128]
    const int tl = target_len[b];   // in [16, 32]

    // ---- extended label sequence: blank, l1, blank, l2, ..., blank ----
    if (tid < CTC_L) {
        ext[tid] = (tid & 1) ? targets[b * CTC_S + (tid >> 1)] : 0;
    }
    if (tid == 0) ext[64] = 0;      // blockDim==64: fold s=64 onto thread 0
    __syncthreads();
    if (tid < CTC_L) {
        skip[tid] = (unsigned char)((tid >= 2) && (tid & 1) && (ext[tid] != ext[tid - 2]));
    }
    if (tid == 0) skip[64] = 0;
    __syncthreads();

    // ---- pre-gather emissions for t < Tb into LDS ----
    const float* lp_b = log_probs + (size_t)b * (size_t)C;
    const size_t row_stride = (size_t)B * (size_t)C;
    const int n = Tb * CTC_L;
    for (int idx = tid; idx < n; idx += (int)blockDim.x) {
        const int t = idx / CTC_L;
        const int s = idx - t * CTC_L;
        const float* src = lp_b + (size_t)t * row_stride + ext[s];
#if CTC_ASYNC_GATHER
        ctc_async_gather_b32(src, &emit[idx]);            // ASYNCcnt-tracked DMA to LDS
#else
        const int idx2 = idx + (int)blockDim.x;           // prefetch next strip
        if (idx2 < n) {
            const int t2 = idx2 / CTC_L;
            const int s2 = idx2 - t2 * CTC_L;
            __builtin_prefetch(lp_b + (size_t)t2 * row_stride + ext[s2], 0, 3);
        }
        emit[idx] = *src;
#endif
    }
#if CTC_ASYNC_GATHER
    ctc_wait_async0();   // drain this wave's async-to-LDS transfers
#endif

    // ---- init alpha at t=0: only s=0 and s=1 reachable ----
    if (tid < 2) {  // left NEG pads for both buffers (never rewritten)
        abuf[0][tid] = CTC_NEG;
        abuf[1][tid] = CTC_NEG;
    }
    __syncthreads();  // emissions + pads visible to all waves
    abuf[0][tid + 2] = (tid < 2) ? emit[tid] : CTC_NEG;
    if (tid == 0) abuf[0][64 + 2] = CTC_NEG;
    __syncthreads();

    // ---- DP over time: read cur / write cur^1, one barrier per step ----
    int cur = 0;
    for (int t = 1; t < Tb; ++t) {
        const float* A = abuf[cur];
        float*       Nw = abuf[cur ^ 1];
        {
            const float a0 = A[tid + 2];
            const float a1 = A[tid + 1];
            const float a2 = skip[tid] ? A[tid] : CTC_NEG;
            Nw[tid + 2] = ctc_lae(ctc_lae(a0, a1), a2) + emit[t * CTC_L + tid];
        }
        if (tid == 0) {  // folded state s = 64
            const int s = 64;
            const float a0 = A[s + 2];
            const float a1 = A[s + 1];
            const float a2 = CTC_NEG;  // s=64 is even: no skip
            Nw[s + 2] = ctc_lae(ctc_lae(a0, a1), a2) + emit[t * CTC_L + s];
        }
        __syncthreads();
        cur ^= 1;
    }

    // ---- finalize: ll = logaddexp(alpha[2*tl], alpha[2*tl-1]) at t = Tb-1 ----
    if (tid == 0) {
        const float aEnd1 = abuf[cur][2 * tl + 2];       // final blank
        const float aEnd0 = abuf[cur][2 * tl - 1 + 2];   // final label
        float loss = -ctc_lae(aEnd1, aEnd0);
        if (loss > 1e29f) loss = 0.0f;                   // zero_infinity
        per_b[b] = loss / (float)tl;                     // reduction='mean' pre-divide
    }
}

__global__ __launch_bounds__(64)
void ctc_reduce_kernel(const float* __restrict__ per_b, float* __restrict__ out, int B)
{
    __shared__ float s[64];
    const int tid = threadIdx.x;
    s[tid] = (tid < B) ? per_b[tid] : 0.0f;
    __syncthreads();
    #pragma unroll
    for (int off = 32; off > 0; off >>= 1) {
        if (tid < off) s[tid] += s[tid + off];
        __syncthreads();
    }
    if (tid == 0) out[0] = s[0] / (float)B;
}

extern "C" void kernel_launch(void* const* d_in, const int* in_sizes, int n_in,
                              void* d_out, int out_size, void* d_ws, size_t ws_size,
                              hipStream_t stream) {
    const float* log_probs  = (const float*)d_in[0];
    const int*   targets    = (const int*)  d_in[1];
    const int*   input_len  = (const int*)  d_in[2];
    const int*   target_len = (const int*)  d_in[3];
    float*       out        = (float*)d_out;
    float*       per_b      = (float*)d_ws;   // B floats of scratch

    const int B = in_sizes[2];                               // 64
    const int T = CTC_TMAX;                                  // 128 (fixed by problem)
    const int C = (int)(in_sizes[0] / ((long long)T * B));   // 6625

    ctc_alpha_kernel<<<B, 64, 0, stream>>>(log_probs, targets, input_len,
                                           target_len, per_b, T, B, C);
    ctc_reduce_kernel<<<1, 64, 0, stream>>>(per_b, out, B);
}